// PolynomialG_8564164788357
// MI455X (gfx1250) — compile-verified
//
#include <hip/hip_runtime.h>
#include <hip/hip_bf16.h>

// ---------------------------------------------------------------------------
// Second-order scan (DEGREE=2, N=5) for MI455X / gfx1250 (wave32).
//
//   h_{t+1} = f(x_t,h_t)                 (sequential scalar chain, kernel k1)
//   J_{t+1} = a_t + b_t * J_t            (affine scan, kernels k2/k3/k4)
//   H_{t+1} = C_t + b_t * H_t            (affine scan, kernels k5/k6/k7)
//
// k7 maps 16 recurrence steps onto V_WMMA_F32_16X16X4_F32:
//   H_local = L * C   with L[r][s] = prod_{u=s+1..r} b_u  (16x16 lower-tri)
//   C is 16 steps x 32 cols (25 Hessian components + padding) -> 2 D-tiles,
//   K=16 accumulated as 4x K=4 WMMA ops per tile (8 wmma / 16 steps).
// L is built cooperatively in LDS (one running product per row) so the WMMA
// operand fetch is branch-free LDS loads — no divergent per-element loops.
// ---------------------------------------------------------------------------

typedef __attribute__((ext_vector_type(2))) float v2f;
typedef __attribute__((ext_vector_type(8))) float v8f;

struct Coefs {
  float w0, w1, w2, w3, w4;      // tanh(params): w_x1,w_x2,w_h1,w_h2,w_c
  float dt0, dt1, dt2, dt3, dt4; // 1-w^2
  float d20, d21, d22, d23, d24; // -2*w*(1-w^2)
};

__device__ inline Coefs make_coefs(const float* __restrict__ p) {
  Coefs c;
  float w;
  w = tanhf(p[0]); c.w0 = w; c.dt0 = 1.f - w * w; c.d20 = -2.f * w * c.dt0;
  w = tanhf(p[1]); c.w1 = w; c.dt1 = 1.f - w * w; c.d21 = -2.f * w * c.dt1;
  w = tanhf(p[2]); c.w2 = w; c.dt2 = 1.f - w * w; c.d22 = -2.f * w * c.dt2;
  w = tanhf(p[3]); c.w3 = w; c.dt3 = 1.f - w * w; c.d23 = -2.f * w * c.dt3;
  w = tanhf(p[4]); c.w4 = w; c.dt4 = 1.f - w * w; c.d24 = -2.f * w * c.dt4;
  return c;
}

__device__ inline float bval(const Coefs& c, float x, float h) {
  return c.w2 + c.w3 * h + c.w4 * x; // df/dh
}
__device__ inline float acomp(const Coefs& c, int n, float x, float h) {
  if (n == 0) return x * c.dt0;
  if (n == 1) return 0.5f * x * x * c.dt1;
  if (n == 2) return h * c.dt2;
  if (n == 3) return 0.5f * h * h * c.dt3;
  return x * h * c.dt4;
}
__device__ inline float vcomp(const Coefs& c, int n, float x, float h) {
  if (n == 0) return 0.f;
  if (n == 1) return 0.f;
  if (n == 2) return c.dt2;
  if (n == 3) return h * c.dt3;
  return x * c.dt4;
}
__device__ inline float ddcomp(const Coefs& c, int n, float x, float h) {
  if (n == 0) return x * c.d20;
  if (n == 1) return 0.5f * x * x * c.d21;
  if (n == 2) return h * c.d22;
  if (n == 3) return 0.5f * h * h * c.d23;
  return x * h * c.d24;
}

// ---------------------------------------------------------------------------
// k1: sequential nonlinear h-chain. Writes Z region (= h sequence) of d_out,
// plus the zero initial rows of J and H. global_prefetch_b8 on X ahead.
// ---------------------------------------------------------------------------
__global__ void k1_hscan(const float* __restrict__ X, const float* __restrict__ P,
                         float* __restrict__ out, int T) {
  if (blockIdx.x != 0 || threadIdx.x != 0) return;
  Coefs c = make_coefs(P);
  out[0] = 0.f;
  for (int n = 0; n < 5; ++n)  out[(long)T + n] = 0.f;   // J[0,:] = 0
  for (int e = 0; e < 25; ++e) out[6L * T + e]  = 0.f;   // H[0,:,:] = 0
  float h = 0.f;
  long TS = (long)T - 1;
  for (long t = 0; t < TS; ++t) {
    __builtin_prefetch(X + t + 1024, 0, 0);  // -> global_prefetch_b8
    float x = X[t];
    h = c.w0 * x + c.w1 * (0.5f * x * x) + c.w2 * h + c.w3 * (0.5f * h * h)
        + c.w4 * x * h;
    out[t + 1] = h;
  }
}

// ---------------------------------------------------------------------------
// k2: per-chunk J-affine reduction. One wave32 per chunk of S steps; each
// lane folds S/32 steps into an affine (A[5],B), then a shfl_up compose scan.
// ---------------------------------------------------------------------------
__global__ void k2_jreduce(const float* __restrict__ X, const float* __restrict__ P,
                           const float* __restrict__ out, float* __restrict__ cJA,
                           float* __restrict__ cB, int T, int S) {
  Coefs cf = make_coefs(P);
  long TS = (long)T - 1;
  long cbase = (long)blockIdx.x * S;
  long cend = cbase + S; if (cend > TS) cend = TS;
  int l = threadIdx.x;
  int sub = S / 32;
  long t0 = cbase + (long)l * sub;
  long t1 = t0 + sub; if (t1 > cend) t1 = cend;
  float A0 = 0.f, A1 = 0.f, A2 = 0.f, A3 = 0.f, A4 = 0.f, B = 1.f;
  for (long t = t0; t < t1; ++t) {
    float x = X[t], h = out[t];
    float b = bval(cf, x, h);
    A0 = x * cf.dt0 + b * A0;
    A1 = 0.5f * x * x * cf.dt1 + b * A1;
    A2 = h * cf.dt2 + b * A2;
    A3 = 0.5f * h * h * cf.dt3 + b * A3;
    A4 = x * h * cf.dt4 + b * A4;
    B *= b;
  }
  for (int off = 1; off < 32; off <<= 1) {
    float pA0 = __shfl_up(A0, off, 32), pA1 = __shfl_up(A1, off, 32);
    float pA2 = __shfl_up(A2, off, 32), pA3 = __shfl_up(A3, off, 32);
    float pA4 = __shfl_up(A4, off, 32), pB  = __shfl_up(B,  off, 32);
    if (l >= off) {
      A0 += B * pA0; A1 += B * pA1; A2 += B * pA2; A3 += B * pA3; A4 += B * pA4;
      B *= pB;
    }
  }
  if (l == 31) {
    long c = blockIdx.x;
    cJA[c * 5 + 0] = A0; cJA[c * 5 + 1] = A1; cJA[c * 5 + 2] = A2;
    cJA[c * 5 + 3] = A3; cJA[c * 5 + 4] = A4;
    cB[c] = B;
  }
}

// k3: exclusive scan over chunk affines -> J value at each chunk start.
__global__ void k3_jscan(const float* __restrict__ cJA, const float* __restrict__ cB,
                         float* __restrict__ Jst, int NC) {
  if (blockIdx.x != 0 || threadIdx.x != 0) return;
  float J0 = 0.f, J1 = 0.f, J2 = 0.f, J3 = 0.f, J4 = 0.f;
  for (int c = 0; c < NC; ++c) {
    Jst[c * 5 + 0] = J0; Jst[c * 5 + 1] = J1; Jst[c * 5 + 2] = J2;
    Jst[c * 5 + 3] = J3; Jst[c * 5 + 4] = J4;
    float B = cB[c];
    J0 = cJA[c * 5 + 0] + B * J0; J1 = cJA[c * 5 + 1] + B * J1;
    J2 = cJA[c * 5 + 2] + B * J2; J3 = cJA[c * 5 + 3] + B * J3;
    J4 = cJA[c * 5 + 4] + B * J4;
  }
}

// k4: per-chunk J write (lanes 0..4 each own one of the 5 J components).
__global__ void k4_jwrite(const float* __restrict__ X, const float* __restrict__ P,
                          float* __restrict__ out, const float* __restrict__ Jst,
                          int T, int S) {
  Coefs cf = make_coefs(P);
  long TS = (long)T - 1;
  long cbase = (long)blockIdx.x * S;
  long cend = cbase + S; if (cend > TS) cend = TS;
  int n = threadIdx.x;
  float J = (n < 5) ? Jst[(long)blockIdx.x * 5 + n] : 0.f;
  for (long t = cbase; t < cend; ++t) {
    float x = X[t], h = out[t];
    float b = bval(cf, x, h);
    float a = acomp(cf, n < 5 ? n : 0, x, h);
    J = a + b * J;
    if (n < 5) out[(long)T + (t + 1) * 5 + n] = J;
  }
}

// k5: per-chunk H-affine reduction. Lanes 0..24 each own component (i,j);
// C_t recomputed from (x_t, h_t, J_t) with J read from d_out.
__global__ void k5_hreduce(const float* __restrict__ X, const float* __restrict__ P,
                           const float* __restrict__ out, float* __restrict__ cHC,
                           int T, int S) {
  Coefs cf = make_coefs(P);
  long TS = (long)T - 1;
  long cbase = (long)blockIdx.x * S;
  long cend = cbase + S; if (cend > TS) cend = TS;
  int e = threadIdx.x;
  int i = (e < 25) ? e / 5 : 0;
  int j = (e < 25) ? e % 5 : 0;
  float Cacc = 0.f;
  for (long t = cbase; t < cend; ++t) {
    float x = X[t], h = out[t];
    float b = bval(cf, x, h);
    float Ji = out[(long)T + t * 5 + i];
    float Jj = out[(long)T + t * 5 + j];
    float Vi = vcomp(cf, i, x, h), Vj = vcomp(cf, j, x, h);
    float dd = (i == j) ? ddcomp(cf, i, x, h) : 0.f;
    float C = dd + Vi * Jj + Ji * Vj + cf.w3 * Ji * Jj;  // f_hh == w_h2
    Cacc = C + b * Cacc;
  }
  if (e < 25) cHC[(long)blockIdx.x * 25 + e] = Cacc;
}

// k6: exclusive scan over chunk H-affines -> H at each chunk start.
__global__ void k6_hscan(const float* __restrict__ cHC, const float* __restrict__ cB,
                         float* __restrict__ Hst, int NC) {
  int e = threadIdx.x;
  if (blockIdx.x != 0 || e >= 25) return;
  float Hv = 0.f;
  for (int c = 0; c < NC; ++c) {
    Hst[(long)c * 25 + e] = Hv;
    Hv = cHC[(long)c * 25 + e] + cB[c] * Hv;
  }
}

// ---------------------------------------------------------------------------
// k7: per-chunk H writer using V_WMMA_F32_16X16X4_F32.
// Per tile of 16 steps:  D(16x16, two tiles for 32 cols) = L(16x16) x C(16x32),
// K=16 as 4x K=4 wmma; then D[r] += p[r]*H_carry; row 15 -> next carry.
// A-operand layout per ISA: VGPR0 = K{+0 lanes0-15, +2 lanes16-31}, VGPR1 = K{+1,+3}.
// L is materialized in LDS: lane m fills row m with one running product, so
// operand gathers are branch-free LDS loads (row stride 17 to avoid bank conflicts).
// ---------------------------------------------------------------------------
__global__ void k7_hwrite(const float* __restrict__ X, const float* __restrict__ P,
                          float* __restrict__ out, const float* __restrict__ Hst,
                          int T, int S) {
  __shared__ float bsh[16], psh[16], xs[16], hs[16];
  __shared__ float Jsh[16][5];
  __shared__ float Cm[16][32];
  __shared__ float Lsh[16][17];   // [row m][col s], padded stride
  __shared__ float Hc[32];
  Coefs cf = make_coefs(P);
  long TS = (long)T - 1;
  long cbase = (long)blockIdx.x * S;
  long cend = cbase + S; if (cend > TS) cend = TS;
  int lane = threadIdx.x;
  int m = lane & 15, kh = lane >> 4;
  Hc[lane] = (lane < 25) ? Hst[(long)blockIdx.x * 25 + lane] : 0.f;
  __syncthreads();

  for (long tb = cbase; tb < cend; tb += 16) {
    if (lane < 16) {
      long t = tb + lane;
      float x = 0.f, h = 0.f, b = 0.f;
      if (t < cend) { x = X[t]; h = out[t]; b = bval(cf, x, h); }
      xs[lane] = x; hs[lane] = h; bsh[lane] = b;
    }
    for (int idx = lane; idx < 80; idx += 32) {
      int s = idx / 5, ii = idx % 5;
      long t = tb + s;
      Jsh[s][ii] = (t < cend) ? out[(long)T + t * 5 + ii] : 0.f;
    }
    __syncthreads();

    // Build L (lower-triangular prefix products) and carry prefixes p[].
    // Lane mm owns row mm: L[mm][mm]=1; L[mm][s] = L[mm][s+1]*b[s+1].
    if (lane < 16) {
      int mm = lane;
      float acc = 1.f;
      Lsh[mm][mm] = 1.f;
      for (int s = mm - 1; s >= 0; --s) { acc *= bsh[s + 1]; Lsh[mm][s] = acc; }
      for (int s = mm + 1; s < 16; ++s) Lsh[mm][s] = 0.f;
      psh[mm] = acc * bsh[0];        // p[mm] = L[mm][0]*b[0] = prod_{0..mm} b
    }
    for (int idx = lane; idx < 512; idx += 32) {  // C matrix 16x32 into LDS
      int s = idx >> 5, e = idx & 31;
      float cv = 0.f;
      if (e < 25 && (tb + s) < cend) {
        int i = e / 5, j = e % 5;
        float x = xs[s], h = hs[s];
        float Ji = Jsh[s][i], Jj = Jsh[s][j];
        float Vi = vcomp(cf, i, x, h), Vj = vcomp(cf, j, x, h);
        float dd = (i == j) ? ddcomp(cf, i, x, h) : 0.f;
        cv = dd + Vi * Jj + Ji * Vj + cf.w3 * Ji * Jj;
      }
      Cm[s][e] = cv;
    }
    __syncthreads();

    v8f d0 = {0.f, 0.f, 0.f, 0.f, 0.f, 0.f, 0.f, 0.f};
    v8f d1 = {0.f, 0.f, 0.f, 0.f, 0.f, 0.f, 0.f, 0.f};
#pragma unroll
    for (int k0 = 0; k0 < 16; k0 += 4) {   // K accumulation, uniform flow
      int sA = k0 + 2 * kh;
      v2f a, b0, b1;
      a.x = Lsh[m][sA];
      a.y = Lsh[m][sA + 1];
      b0.x = Cm[sA][m];      b0.y = Cm[sA + 1][m];
      b1.x = Cm[sA][16 + m]; b1.y = Cm[sA + 1][16 + m];
      d0 = __builtin_amdgcn_wmma_f32_16x16x4_f32(false, a, false, b0,
                                                 (short)0, d0, false, false);
      d1 = __builtin_amdgcn_wmma_f32_16x16x4_f32(false, a, false, b1,
                                                 (short)0, d1, false, false);
    }
    float hc0 = Hc[m];        // carry comps 0..15
    float hc1 = Hc[16 + m];   // comps 16..24 (25..31 are zero padding)
#pragma unroll
    for (int v = 0; v < 8; ++v) {
      int r = v + 8 * kh;
      float pr = psh[r];
      d0[v] += pr * hc0;
      d1[v] += pr * hc1;
    }
    __syncthreads();          // all lanes done reading old carry
    if (kh == 1) {            // row r=15 lives in v=7 of upper lane half
      Hc[m] = d0[7];
      if (16 + m < 25) Hc[16 + m] = d1[7];
    }
#pragma unroll
    for (int v = 0; v < 8; ++v) {
      int r = v + 8 * kh;
      long tout = tb + 1 + r;
      if (tout <= cend) {
        long base = 6L * T + tout * 25;
        out[base + m] = d0[v];
        if (16 + m < 25) out[base + 16 + m] = d1[v];
      }
    }
    __syncthreads();
  }
}

// ---------------------------------------------------------------------------
extern "C" void kernel_launch(void* const* d_in, const int* in_sizes, int n_in,
                              void* d_out, int out_size, void* d_ws, size_t ws_size,
                              hipStream_t stream) {
  const float* X = (const float*)d_in[0];
  const float* P = (const float*)d_in[1];
  float* out = (float*)d_out;
  float* ws = (float*)d_ws;
  int T = in_sizes[0];
  long TS = (long)T - 1;
  const int S = 1024;                       // steps per chunk (multiple of 16)
  int NC = (TS > 0) ? (int)((TS + S - 1) / S) : 0;

  // ws layout (floats): 61*NC total (~240 KB for T=1e6)
  float* chunkJA = ws;                 // NC*5
  float* chunkB  = ws + 5L * NC;       // NC
  float* Jstart  = ws + 6L * NC;       // NC*5
  float* chunkHC = ws + 11L * NC;      // NC*25
  float* Hstart  = ws + 36L * NC;      // NC*25

  k1_hscan<<<1, 32, 0, stream>>>(X, P, out, T);
  if (NC > 0) {
    k2_jreduce<<<NC, 32, 0, stream>>>(X, P, out, chunkJA, chunkB, T, S);
    k3_jscan<<<1, 32, 0, stream>>>(chunkJA, chunkB, Jstart, NC);
    k4_jwrite<<<NC, 32, 0, stream>>>(X, P, out, Jstart, T, S);
    k5_hreduce<<<NC, 32, 0, stream>>>(X, P, out, chunkHC, T, S);
    k6_hscan<<<1, 32, 0, stream>>>(chunkHC, chunkB, Hstart, NC);
    k7_hwrite<<<NC, 32, 0, stream>>>(X, P, out, Hstart, T, S);
  }
}